// ONNXDGCNNBackbone_64896955842762
// MI455X (gfx1250) — compile-verified
//
#include <hip/hip_runtime.h>
#include <math.h>

typedef float v2f __attribute__((ext_vector_type(2)));
typedef float v8f __attribute__((ext_vector_type(8)));

#define KNN    20
#define SEG    5          // column-range split for the kNN phase (625 tiles = 5 x 125)
#define CTOT   512
#define EPSV   1e-5f
#define SLOPEV 0.2f

// ---------------------------------------------------------------------------
// prep: transpose x (C x N, channel stride N) -> xT (N x C, row major)
//       and compute per-point squared norms xx[n].
// ---------------------------------------------------------------------------
template<int C>
__global__ void __launch_bounds__(256)
prep_kernel(const float* __restrict__ x, size_t bstride,
            float* __restrict__ xT, float* __restrict__ xx, int N)
{
  int n = blockIdx.x * 256 + threadIdx.x;
  int b = blockIdx.y;
  if (n >= N) return;
  const float* xb = x + (size_t)b * bstride;
  float* dst = xT + ((size_t)b * N + n) * C;
  float s = 0.f;
#pragma unroll 8
  for (int c = 0; c < C; ++c) {
    float v = xb[(size_t)c * N + n];
    dst[c] = v;
    s = fmaf(v, v, s);
  }
  xx[(size_t)b * N + n] = s;
}

// ---------------------------------------------------------------------------
// knn (phase 1): one wave per (16-row tile, column segment). Gram matrix via
// V_WMMA_F32_16X16X4_F32, pd = 2*S - xx_i - xx_j staged to LDS, lanes 0..15
// keep a register-resident top-20 list for their segment, then dump the 20
// (value, index) candidates to the candidate buffer.
// ---------------------------------------------------------------------------
template<int C>
__global__ void __launch_bounds__(32)
knn_kernel(const float* __restrict__ xT, const float* __restrict__ xx,
           float* __restrict__ candV, int* __restrict__ candI, int N)
{
  const int b    = blockIdx.z;
  const int seg  = blockIdx.y;
  const int ri   = blockIdx.x;          // row tile index (16 rows)
  const int lane = threadIdx.x;
  const int mh   = lane & 15;           // M (or N) position within tile
  const int kh   = lane >> 4;           // K-half selector
  constexpr int KG = C / 4;             // k-groups of 4 per WMMA

  const float* xTb = xT + (size_t)b * N * C;
  const float* xxb = xx + (size_t)b * N;

  // A fragments for this wave's 16 rows (reused for every column tile).
  // A 16x4 f32 layout: lanes 0-15 hold K=4kg+{0,1}, lanes 16-31 K=4kg+{2,3}.
  v2f afrag[KG];
  const int row = ri * 16 + mh;
  {
    const float* p = xTb + (size_t)row * C + kh * 2;
#pragma unroll
    for (int kg = 0; kg < KG; ++kg) {
      afrag[kg][0] = p[kg * 4 + 0];
      afrag[kg][1] = p[kg * 4 + 1];
    }
  }

  // Row norms for the D-layout rows this lane touches: M = g + 8*kh.
  float xxi[8];
#pragma unroll
  for (int g = 0; g < 8; ++g) xxi[g] = xxb[ri * 16 + g + 8 * kh];

  // Unsorted top-K list (registers; all index loops fully unrolled).
  float val[KNN]; int vid[KNN];
#pragma unroll
  for (int t = 0; t < KNN; ++t) { val[t] = -__builtin_inff(); vid[t] = 0; }
  float minv = -__builtin_inff(); int minp = 0;

  // 20-float row stride: rows 16B-aligned, conflict-free for both the
  // column-strided WMMA-result stores and the b128 row scans.
  __shared__ __align__(16) float tile[16][20];

  const int tps = (N / 16) / SEG;       // tiles per segment
  const int t0  = seg * tps;
  for (int cj = t0; cj < t0 + tps; ++cj) {
    v8f acc = {0.f,0.f,0.f,0.f,0.f,0.f,0.f,0.f};
    const float* pb = xTb + (size_t)(cj * 16 + mh) * C + kh * 2;
#pragma unroll
    for (int kg = 0; kg < KG; ++kg) {
      v2f bfrag;
      bfrag[0] = pb[kg * 4 + 0];
      bfrag[1] = pb[kg * 4 + 1];
      acc = __builtin_amdgcn_wmma_f32_16x16x4_f32(
          false, afrag[kg], false, bfrag, (short)0, acc, false, false);
    }
    // pd = -xx_i + 2*S - xx_j ; D layout: M = g + 8*kh, N = mh
    float xxj = xxb[cj * 16 + mh];
#pragma unroll
    for (int g = 0; g < 8; ++g)
      tile[g + 8 * kh][mh] = 2.f * acc[g] - xxi[g] - xxj;

    __syncthreads();   // single-wave workgroup: barrier is NOP, waits remain

    if (lane < 16) {
      const float4* rowp = (const float4*)&tile[lane][0];
#pragma unroll
      for (int jj = 0; jj < 4; ++jj) {
        float4 q = rowp[jj];
#pragma unroll
        for (int c = 0; c < 4; ++c) {
          float d  = (c == 0) ? q.x : (c == 1) ? q.y : (c == 2) ? q.z : q.w;
          int  col = cj * 16 + jj * 4 + c;
          if (d > minv) {
#pragma unroll
            for (int t = 0; t < KNN; ++t)
              if (t == minp) { val[t] = d; vid[t] = col; }
            minv = val[0]; minp = 0;
#pragma unroll
            for (int t = 1; t < KNN; ++t)
              if (val[t] < minv) { minv = val[t]; minp = t; }
          }
        }
      }
    }
    __syncthreads();
  }

  if (lane < 16) {
    size_t base = ((size_t)b * N + ri * 16 + lane) * (SEG * KNN) + (size_t)seg * KNN;
#pragma unroll
    for (int t = 0; t < KNN; ++t) {
      candV[base + t] = val[t];
      candI[base + t] = vid[t];
    }
  }
}

// ---------------------------------------------------------------------------
// knn (phase 2): merge the SEG per-segment top-20 lists -> final top-20 idx.
// One thread per point, candidates contiguous (b128 loads).
// ---------------------------------------------------------------------------
__global__ void __launch_bounds__(256)
merge_kernel(const float* __restrict__ candV, const int* __restrict__ candI,
             int* __restrict__ idx, int N, int B)
{
  int t = blockIdx.x * 256 + threadIdx.x;
  if (t >= B * N) return;
  const float4* pv = (const float4*)(candV + (size_t)t * (SEG * KNN));
  const int4*   pi = (const int4*)(candI + (size_t)t * (SEG * KNN));

  float val[KNN]; int vid[KNN];
#pragma unroll
  for (int j = 0; j < KNN; ++j) { val[j] = -__builtin_inff(); vid[j] = 0; }
  float minv = -__builtin_inff(); int minp = 0;

  for (int q = 0; q < (SEG * KNN) / 4; ++q) {
    float4 v4 = pv[q];
    int4   i4 = pi[q];
#pragma unroll
    for (int c = 0; c < 4; ++c) {
      float d = (c == 0) ? v4.x : (c == 1) ? v4.y : (c == 2) ? v4.z : v4.w;
      int   i = (c == 0) ? i4.x : (c == 1) ? i4.y : (c == 2) ? i4.z : i4.w;
      if (d > minv) {
#pragma unroll
        for (int u = 0; u < KNN; ++u)
          if (u == minp) { val[u] = d; vid[u] = i; }
        minv = val[0]; minp = 0;
#pragma unroll
        for (int u = 1; u < KNN; ++u)
          if (val[u] < minv) { minv = val[u]; minp = u; }
      }
    }
  }

  int* op = idx + (size_t)t * KNN;
#pragma unroll
  for (int j = 0; j < KNN; ++j) op[j] = vid[j];   // order irrelevant (max over k)
}

// ---------------------------------------------------------------------------
// gemm: G[b][n][0:COUT]      = x[n] . (W_L + W_R)[o]   (center term, GA)
//       G[b][n][COUT:2COUT]  = x[n] . W_L[o]           (neighbor term, GB)
// One wave per 16x16 output tile, fp32 WMMA over the CIN contraction.
// ---------------------------------------------------------------------------
template<int CIN, int COUT>
__global__ void __launch_bounds__(32)
gemm_kernel(const float* __restrict__ xT, const float* __restrict__ w,
            float* __restrict__ G, int N)
{
  const int b    = blockIdx.z;
  const int ri   = blockIdx.x;
  const int ci   = blockIdx.y;          // 16-col tile over 2*COUT
  const int lane = threadIdx.x;
  const int mh   = lane & 15;
  const int kh   = lane >> 4;
  constexpr int KG = CIN / 4;

  const float* xb   = xT + (size_t)b * N * CIN;
  const int    ocol = ci * 16 + mh;           // 0 .. 2*COUT-1
  const bool   isGA = (ci * 16) < COUT;       // uniform per tile (COUT%16==0)
  const int    o    = isGA ? ocol : (ocol - COUT);

  const float* arow = xb + (size_t)(ri * 16 + mh) * CIN + kh * 2;
  const float* wrow = w + (size_t)o * (2 * CIN) + kh * 2;

  v8f acc = {0.f,0.f,0.f,0.f,0.f,0.f,0.f,0.f};
#pragma unroll
  for (int kg = 0; kg < KG; ++kg) {
    v2f a, bb;
    a[0] = arow[kg * 4 + 0];
    a[1] = arow[kg * 4 + 1];
    float b0 = wrow[kg * 4 + 0];
    float b1 = wrow[kg * 4 + 1];
    if (isGA) { b0 += wrow[CIN + kg * 4 + 0]; b1 += wrow[CIN + kg * 4 + 1]; }
    bb[0] = b0; bb[1] = b1;
    acc = __builtin_amdgcn_wmma_f32_16x16x4_f32(
        false, a, false, bb, (short)0, acc, false, false);
  }

  float* Gb = G + (size_t)b * N * (2 * COUT);
#pragma unroll
  for (int g = 0; g < 8; ++g) {
    int n = ri * 16 + g + 8 * kh;
    Gb[(size_t)n * (2 * COUT) + ocol] = acc[g];
  }
}

// ---------------------------------------------------------------------------
// reduce: out[b][base+o][n] = max_k leaky( (GA[n,o]-GB[idx[n,k],o])*sc + sh )
// ---------------------------------------------------------------------------
template<int COUT>
__global__ void __launch_bounds__(256)
reduce_kernel(const float* __restrict__ G, const int* __restrict__ idx,
              const float* __restrict__ gam, const float* __restrict__ bet,
              const float* __restrict__ mu,  const float* __restrict__ var,
              float* __restrict__ out, int N, int B, int ch_base)
{
  long t = (long)blockIdx.x * 256 + threadIdx.x;
  int  o  = (int)(t % COUT);
  long nn = t / COUT;
  int  n  = (int)(nn % N);
  int  b  = (int)(nn / N);
  if (b >= B) return;

  float sc = gam[o] * rsqrtf(var[o] + EPSV);
  float sh = bet[o] - mu[o] * sc;

  const float* Gb = G + (size_t)b * N * (2 * COUT);
  float ga = Gb[(size_t)n * (2 * COUT) + o];
  const int* ip = idx + ((size_t)b * N + n) * KNN;

  float best = -__builtin_inff();
#pragma unroll
  for (int k = 0; k < KNN; ++k) {
    int   j  = ip[k];
    float yb = Gb[(size_t)j * (2 * COUT) + COUT + o];
    float y  = fmaf(ga - yb, sc, sh);
    y = (y >= 0.f) ? y : SLOPEV * y;
    best = fmaxf(best, y);
  }
  out[((size_t)b * CTOT + ch_base + o) * N + n] = best;
}

// ---------------------------------------------------------------------------
// Host-side layer driver
// ---------------------------------------------------------------------------
template<int CIN, int COUT>
static void run_layer(const float* xin, size_t bstride,
                      const float* w, const float* g, const float* bt,
                      const float* m, const float* v,
                      float* out, int ch_base,
                      float* xT, float* xx, int* idx, float* G,
                      float* candV, int* candI,
                      int N, int B, hipStream_t s)
{
  prep_kernel<CIN><<<dim3((N + 255) / 256, B), 256, 0, s>>>(xin, bstride, xT, xx, N);
  knn_kernel<CIN><<<dim3(N / 16, SEG, B), 32, 0, s>>>(xT, xx, candV, candI, N);
  merge_kernel<<<(B * N + 255) / 256, 256, 0, s>>>(candV, candI, idx, N, B);
  gemm_kernel<CIN, COUT><<<dim3(N / 16, (2 * COUT) / 16, B), 32, 0, s>>>(xT, w, G, N);
  long total = (long)B * N * COUT;
  reduce_kernel<COUT><<<(unsigned)((total + 255) / 256), 256, 0, s>>>(
      G, idx, g, bt, m, v, out, N, B, ch_base);
}

extern "C" void kernel_launch(void* const* d_in, const int* in_sizes, int n_in,
                              void* d_out, int out_size, void* d_ws, size_t ws_size,
                              hipStream_t stream)
{
  (void)in_sizes; (void)n_in; (void)out_size; (void)ws_size;
  const int N = 10000, B = 2;

  const float* x = (const float*)d_in[0];
  float* out = (float*)d_out;

  // Workspace layout (~69 MB total)
  float* xT    = (float*)d_ws;                          // B*N*128 floats (max CIN)
  float* xx    = xT + (size_t)B * N * 128;              // B*N floats
  int*   idx   = (int*)(xx + (size_t)B * N);            // B*N*KNN ints
  float* G     = (float*)(idx + (size_t)B * N * KNN);   // B*N*512 floats (max 2*COUT)
  float* candV = G + (size_t)B * N * 512;               // B*N*SEG*KNN floats
  int*   candI = (int*)(candV + (size_t)B * N * SEG * KNN); // B*N*SEG*KNN ints

  const float *w1=(const float*)d_in[1],  *g1=(const float*)d_in[2],
              *b1=(const float*)d_in[3],  *m1=(const float*)d_in[4],  *v1=(const float*)d_in[5];
  const float *w2=(const float*)d_in[6],  *g2=(const float*)d_in[7],
              *b2=(const float*)d_in[8],  *m2=(const float*)d_in[9],  *v2=(const float*)d_in[10];
  const float *w3=(const float*)d_in[11], *g3=(const float*)d_in[12],
              *b3=(const float*)d_in[13], *m3=(const float*)d_in[14], *v3=(const float*)d_in[15];
  const float *w4=(const float*)d_in[16], *g4=(const float*)d_in[17],
              *b4=(const float*)d_in[18], *m4=(const float*)d_in[19], *v4=(const float*)d_in[20];

  // Layer 1: input is x (B,4,N); output -> out channels [0,64)
  run_layer<4, 64>(x, (size_t)4 * N, w1, g1, b1, m1, v1,
                   out, 0, xT, xx, idx, G, candV, candI, N, B, stream);
  // Layer 2: input x1 = out channels [0,64); output -> [64,128)
  run_layer<64, 64>(out, (size_t)CTOT * N, w2, g2, b2, m2, v2,
                    out, 64, xT, xx, idx, G, candV, candI, N, B, stream);
  // Layer 3: input x2 = out channels [64,128); output -> [128,256)
  run_layer<64, 128>(out + (size_t)64 * N, (size_t)CTOT * N, w3, g3, b3, m3, v3,
                     out, 128, xT, xx, idx, G, candV, candI, N, B, stream);
  // Layer 4: input x3 = out channels [128,256); output -> [256,512)
  run_layer<128, 256>(out + (size_t)128 * N, (size_t)CTOT * N, w4, g4, b4, m4, v4,
                      out, 256, xT, xx, idx, G, candV, candI, N, B, stream);
}